// EGAT_69956427317436
// MI455X (gfx1250) — compile-verified
//
#include <hip/hip_runtime.h>

typedef float v2f __attribute__((ext_vector_type(2)));
typedef float v8f __attribute__((ext_vector_type(8)));

// z = alpha * (x[col] @ W.T) segment-summed by row; softmax over axis=1 of an
// [E,1] tensor is identically 1, so alpha == 1 and by linearity:
//   z = segment_sum(x[col], row) @ W.T
// Stage 1: zero S. Stage 2: atomic scatter-add of x[col] rows into S[row].
// Stage 3: z = S @ W.T via V_WMMA_F32_16X16X4_F32 (one wave per 16x16 tile).

// ---------------- Stage 1: zero accumulator S [N,64] ----------------
__global__ void egat_zero(float4* __restrict__ S4, int n4) {
  int i = blockIdx.x * blockDim.x + threadIdx.x;
  if (i < n4) S4[i] = make_float4(0.f, 0.f, 0.f, 0.f);
}

// ---------------- Stage 2: S[row[e],:] += x[col[e],:] ----------------
// 16 threads per edge, 4 channels each (b128 gather + 4x global_atomic_add_f32).
// x (25.6 MB) and S (25.6 MB) both fit in the 192 MB L2, so gathers/atomics
// stay on-chip.
__global__ void egat_scatter(const float* __restrict__ x,
                             const int* __restrict__ row,
                             const int* __restrict__ col,
                             float* __restrict__ S, int E) {
  int tid = blockIdx.x * blockDim.x + threadIdx.x;
  int e = tid >> 4;
  if (e >= E) return;
  int c0 = (tid & 15) << 2;
  int r = row[e];
  int c = col[e];
  const float4 v = *(const float4*)(x + (size_t)c * 64 + c0);
  float* dst = S + (size_t)r * 64 + c0;
  atomicAdd(dst + 0, v.x);
  atomicAdd(dst + 1, v.y);
  atomicAdd(dst + 2, v.z);
  atomicAdd(dst + 3, v.w);
}

// ---------------- Stage 3: z = S @ W.T via WMMA f32 16x16x4 ----------------
// Block = 128 threads = 4 waves; block b owns row tile R=16*b, wave w owns
// column tile C=16*w. Per ISA 7.12.2:
//   A 16x4 f32 : lanes 0-15 -> {K=0,K=1}, lanes 16-31 -> {K=2,K=3}, M=lane&15
//   B 4x16 f32 : lanes 0-15 -> {K=0,K=1}, lanes 16-31 -> {K=2,K=3}, N=lane&15
//   C/D 16x16  : VGPR r -> M=r (lanes 0-15) / M=r+8 (lanes 16-31), N=lane&15
__global__ void egat_gemm(const float* __restrict__ S,
                          const float* __restrict__ W,
                          float* __restrict__ z) {
  const int lane = threadIdx.x & 31;
  const int wave = threadIdx.x >> 5;   // 0..3 -> column tile
  const int R    = blockIdx.x * 16;    // row tile base
  const int C    = wave * 16;          // col tile base
  const int half = lane >> 4;          // lane group
  const int m    = lane & 15;

  const float* __restrict__ srow = S + (size_t)(R + m) * 64 + 2 * half;
  const float* __restrict__ wrow = W + (size_t)(C + m) * 64 + 2 * half;

  v8f acc = {};
#pragma unroll
  for (int kb = 0; kb < 64; kb += 4) {
    v2f a, b;
    a.x = srow[kb + 0];  // A[m, kb + 2*half + 0]
    a.y = srow[kb + 1];  // A[m, kb + 2*half + 1]
    b.x = wrow[kb + 0];  // B[kb + 2*half + 0, m] == W[C+m, kb + 2*half + 0]
    b.y = wrow[kb + 1];
    // (neg_a, A, neg_b, B, c_mod, C, reuse_a, reuse_b)
    acc = __builtin_amdgcn_wmma_f32_16x16x4_f32(false, a, false, b,
                                                (short)0, acc, false, false);
  }

  float* zt = z + (size_t)R * 64 + C + m;
#pragma unroll
  for (int r = 0; r < 8; ++r) {
    zt[(size_t)(r + half * 8) * 64] = acc[r];
  }
}

// Scalar tail for N % 16 != 0 (not hit for N=100000, kept for generality).
__global__ void egat_gemm_tail(const float* __restrict__ S,
                               const float* __restrict__ W,
                               float* __restrict__ z, int startRow, int N) {
  int r = startRow + blockIdx.x;
  if (r >= N) return;
  int o = threadIdx.x;  // 64 output channels
  float acc = 0.f;
#pragma unroll 8
  for (int k = 0; k < 64; ++k) acc += S[(size_t)r * 64 + k] * W[(size_t)o * 64 + k];
  z[(size_t)r * 64 + o] = acc;
}

extern "C" void kernel_launch(void* const* d_in, const int* in_sizes, int n_in,
                              void* d_out, int out_size, void* d_ws, size_t ws_size,
                              hipStream_t stream) {
  // Inputs (setup_inputs order): x, edge_index, edge_attr, W, We, Wa.
  // edge_attr/We/Wa are mathematically dead (softmax over singleton axis == 1).
  const float* x  = (const float*)d_in[0];
  const int*   ei = (const int*)d_in[1];   // [2, E] int32
  const float* W  = (const float*)d_in[3]; // [64, 64]
  float* z = (float*)d_out;
  float* S = (float*)d_ws;                 // [N, 64] accumulator

  const int N = in_sizes[0] / 64;
  const int E = in_sizes[1] / 2;

  // Stage 1: zero S.
  const int n4 = N * 16;  // float4 count
  egat_zero<<<(n4 + 255) / 256, 256, 0, stream>>>((float4*)S, n4);

  // Stage 2: scatter-add (16 threads per edge).
  const long long total = (long long)E * 16;
  egat_scatter<<<(int)((total + 255) / 256), 256, 0, stream>>>(
      x, ei /*row*/, ei + E /*col*/, S, E);

  // Stage 3: WMMA GEMM over full 16-row tiles, scalar tail otherwise.
  const int fullTiles = N / 16;
  if (fullTiles > 0)
    egat_gemm<<<fullTiles, 128, 0, stream>>>(S, W, z);
  const int tail = N - fullTiles * 16;
  if (tail > 0)
    egat_gemm_tail<<<tail, 64, 0, stream>>>(S, W, z, fullTiles * 16, N);
}